// ModelNew_10548439679732
// MI455X (gfx1250) — compile-verified
//
#include <hip/hip_runtime.h>

// Masked inclusive row-cumsum, 2048 rows x 32768 f32.
// One workgroup (1024 threads = 32 wave32) per row; 4 sequential tiles of 8192.
// Per-wave 256-element chunk scanned on the matrix pipe with
// V_WMMA_F32_16X16X4_F32:  S = X*U + (Lstrict*X)*J   (all f32, no precision loss).

typedef float v2f __attribute__((ext_vector_type(2)));
typedef float v8f __attribute__((ext_vector_type(8)));

#define ROWS    2048
#define NCOLS   32768
#define TILE    8192
#define NTILES  (NCOLS / TILE)   // 4
#define THREADS 1024
#define NWAVES  32
#define CHUNK   256              // per-wave elements (16x16 f32 matrix)

__device__ __forceinline__ float wave_sum32(float v) {
#pragma unroll
    for (int o = 16; o > 0; o >>= 1) v += __shfl_xor(v, o, 32);
    return v;
}

__global__ __launch_bounds__(THREADS)
void masked_cumsum_wmma(const float* __restrict__ x,
                        const unsigned char* __restrict__ mask,
                        float* __restrict__ out) {
    __shared__ float stage[TILE];     // masked values, one 256-chunk per wave
    __shared__ float tot[NWAVES];     // per-chunk totals
    __shared__ float carry;           // running row carry across tiles

    const int tid  = threadIdx.x;
    const int lane = tid & 31;
    const int wave = tid >> 5;
    const int lo   = lane & 15;       // 0..15
    const int hi   = lane >> 4;       // 0..1

    const size_t rowBase = (size_t)blockIdx.x * NCOLS;
    const float*         xRow = x    + rowBase;
    const unsigned char* mRow = mask + rowBase;
    float*               oRow = out  + rowBase;

    // Constant WMMA operands (lane-dependent only).
    // A layout (16x4 f32):  lane holds Mat[lo][k0+v], k0 = 4*kb + 2*hi, v=0..1
    // B layout (4x16 f32):  lane holds Mat[k0+v][lo]
    v2f Ub[4];   // U[k][n] = (k <= n)  -> B operand
    v2f La[4];   // Lstrict[m][k] = (k < m) -> A operand
#pragma unroll
    for (int kb = 0; kb < 4; ++kb) {
        const int k0 = 4 * kb + 2 * hi;
        Ub[kb].x = (k0     <= lo) ? 1.0f : 0.0f;
        Ub[kb].y = (k0 + 1 <= lo) ? 1.0f : 0.0f;
        La[kb].x = (k0     <  lo) ? 1.0f : 0.0f;
        La[kb].y = (k0 + 1 <  lo) ? 1.0f : 0.0f;
    }
    v2f Jb; Jb.x = 1.0f; Jb.y = 1.0f;   // all-ones B operand

    if (tid == 0) carry = 0.0f;

    for (int t = 0; t < NTILES; ++t) {
        const int tileBase = t * TILE;

        // ---- stage masked values into LDS (each wave stages its own chunk) ----
        {
            const int e = tid * 8;
            float4 a = *(const float4*)(xRow + tileBase + e);
            float4 b = *(const float4*)(xRow + tileBase + e + 4);
            uchar4 ma = *(const uchar4*)(mRow + tileBase + e);
            uchar4 mb = *(const uchar4*)(mRow + tileBase + e + 4);
            float4 va, vb;
            va.x = ma.x ? a.x : 0.0f;  va.y = ma.y ? a.y : 0.0f;
            va.z = ma.z ? a.z : 0.0f;  va.w = ma.w ? a.w : 0.0f;
            vb.x = mb.x ? b.x : 0.0f;  vb.y = mb.y ? b.y : 0.0f;
            vb.z = mb.z ? b.z : 0.0f;  vb.w = mb.w ? b.w : 0.0f;
            *(float4*)(stage + e)     = va;
            *(float4*)(stage + e + 4) = vb;
            if (t + 1 < NTILES) {   // global_prefetch_b8 of next tile
                __builtin_prefetch(xRow + tileBase + TILE + e, 0, 0);
                __builtin_prefetch(mRow + tileBase + TILE + e, 0, 0);
            }
        }
        __syncthreads();   // B1 (also publishes carry from previous tile)

        float* chunk = stage + wave * CHUNK;   // wave-private 16x16 matrix X

        // Load X in both A and B layouts from LDS.
        v2f Xa[4], Xb[4];
#pragma unroll
        for (int kb = 0; kb < 4; ++kb) {
            const int k0 = 4 * kb + 2 * hi;
            Xa[kb] = *(const v2f*)(chunk + 16 * lo + k0);      // contiguous pair
            Xb[kb].x = chunk[16 * k0 + lo];
            Xb[kb].y = chunk[16 * (k0 + 1) + lo];
        }

        // Y = X * U   (row-wise inclusive prefix)
        v8f y = {};
#pragma unroll
        for (int kb = 0; kb < 4; ++kb)
            y = __builtin_amdgcn_wmma_f32_16x16x4_f32(false, Xa[kb], false, Ub[kb],
                                                      (short)0, y, false, false);

        // T = Lstrict * X  (column sums of preceding rows)
        v8f tm = {};
#pragma unroll
        for (int kb = 0; kb < 4; ++kb)
            tm = __builtin_amdgcn_wmma_f32_16x16x4_f32(false, La[kb], false, Xb[kb],
                                                       (short)0, tm, false, false);

        // Round-trip T through wave-private LDS: D layout -> row-major -> A layout.
        // D layout: lane holds (M = v + 8*hi, N = lo), v = 0..7.
#pragma unroll
        for (int v = 0; v < 8; ++v)
            chunk[16 * v + 128 * hi + lo] = tm[v];
        v2f Ta[4];
#pragma unroll
        for (int kb = 0; kb < 4; ++kb) {
            const int k0 = 4 * kb + 2 * hi;
            Ta[kb] = *(const v2f*)(chunk + 16 * lo + k0);
        }

        // S = T * J + Y   (broadcast preceding-rows total, add row prefix)
        v8f s = y;
#pragma unroll
        for (int kb = 0; kb < 4; ++kb)
            s = __builtin_amdgcn_wmma_f32_16x16x4_f32(false, Ta[kb], false, Jb,
                                                      (short)0, s, false, false);

        // Chunk total lives at M=15,N=15 -> lane 31, s[7].
        if (lane == 31) tot[wave] = s[7];
        __syncthreads();   // B2

        // Exclusive prefix of chunk totals across the 32 waves + row carry.
        const float tv   = tot[lane];
        const float full = wave_sum32(tv);
        const float pre  = wave_sum32((lane < wave) ? tv : 0.0f);
        const float off  = carry + pre;

        float* oChunk = oRow + tileBase + wave * CHUNK;
#pragma unroll
        for (int v = 0; v < 8; ++v)
            oChunk[16 * v + 128 * hi + lo] = s[v] + off;

        __syncthreads();   // B3 (all reads of carry done)
        if (tid == 0) carry += full;
        // next tile's B1 publishes the updated carry
    }
}

extern "C" void kernel_launch(void* const* d_in, const int* in_sizes, int n_in,
                              void* d_out, int out_size, void* d_ws, size_t ws_size,
                              hipStream_t stream) {
    (void)in_sizes; (void)n_in; (void)d_ws; (void)ws_size; (void)out_size;
    const float*         x    = (const float*)d_in[0];
    const unsigned char* mask = (const unsigned char*)d_in[1];  // jax bool_ = 1 byte
    float*               out  = (float*)d_out;
    masked_cumsum_wmma<<<ROWS, THREADS, 0, stream>>>(x, mask, out);
}